// ModelNew_58798102282514
// MI455X (gfx1250) — compile-verified
//
#include <hip/hip_runtime.h>

typedef __attribute__((ext_vector_type(2))) float v2f;
typedef __attribute__((ext_vector_type(8))) float v8f;

#define BATCH 64
#define CIN   64
#define COUT  64
#define HIN   128
#define WIN   128
#define PLANE (HIN*WIN)        // 16384 elements per (b,c) plane
#define KDIM  256              // 4 groups x 64 channels
#define INV_AREA (1.0f/65536.0f)
#define MULTV 0.5f

// ---------------------------------------------------------------------------
// Kernel 1: per-(b,c) plane reduction -> A[64][256]
// A[b][c]      =  T   (full plane sum)
// A[b][64+c]   = -R0  (row h==0 sum)
// A[b][128+c]  = -C0  (col w==0 sum)
// A[b][192+c]  =  X00 (x[b,c,0,0])
// Memory-bound: 268 MB streamed, fully coalesced float4 loads.
// ---------------------------------------------------------------------------
__global__ void __launch_bounds__(256)
reduce_planes(const float* __restrict__ x, float* __restrict__ A) {
    const int plane = blockIdx.x;          // b*64 + c
    const int b = plane >> 6;
    const int c = plane & 63;
    const int t = threadIdx.x;

    const float4* p = (const float4*)(x + (size_t)plane * PLANE);

    float s_t = 0.f, s_r = 0.f, s_c = 0.f, s_x = 0.f;
#pragma unroll
    for (int i = 0; i < 16; ++i) {
        const int idx = t + 256 * i;       // float4 index, 0..4095, coalesced
        const float4 v = p[idx];
        const float s = v.x + v.y + v.z + v.w;
        s_t += s;
        if (idx < 32)        s_r += s;     // h == 0 (first 128 floats)
        if ((idx & 31) == 0) s_c += v.x;   // w == 0 (idx = 32*h, elem 0)
        if (idx == 0)        s_x  = v.x;   // x[b,c,0,0]
    }

    __shared__ float sm0[256], sm1[256], sm2[256], sm3[256];
    sm0[t] = s_t; sm1[t] = s_r; sm2[t] = s_c; sm3[t] = s_x;
    __syncthreads();
    for (int s = 128; s > 0; s >>= 1) {
        if (t < s) {
            sm0[t] += sm0[t + s];
            sm1[t] += sm1[t + s];
            sm2[t] += sm2[t + s];
            sm3[t] += sm3[t + s];
        }
        __syncthreads();
    }
    if (t == 0) {
        float* Arow = A + (size_t)b * KDIM;
        Arow[c]       =  sm0[0];
        Arow[64 + c]  = -sm1[0];
        Arow[128 + c] = -sm2[0];
        Arow[192 + c] =  sm3[0];
    }
}

// ---------------------------------------------------------------------------
// Kernel 2: weight prep -> B[256][64]
// weight layout [Cin][Cout][3][3].
// B[c][o]=sum_all, B[64+c][o]=sum(kh=0 row), B[128+c][o]=sum(kw=0 col),
// B[192+c][o]=w[0][0].
// ---------------------------------------------------------------------------
__global__ void __launch_bounds__(256)
prep_weights(const float* __restrict__ w, float* __restrict__ Bm) {
    const int t = blockIdx.x * blockDim.x + threadIdx.x;  // 0..4095 = c*64+o
    if (t >= CIN * COUT) return;
    const int c = t >> 6;
    const int o = t & 63;
    const float* wp = w + (size_t)t * 9;
    const float w00 = wp[0], w01 = wp[1], w02 = wp[2];
    const float w10 = wp[3],              w12 = wp[5];
    const float w20 = wp[6], w21 = wp[7], w22 = wp[8];
    const float w11 = wp[4];
    const float wsum = w00 + w01 + w02 + w10 + w11 + w12 + w20 + w21 + w22;
    const float wk0  = w00 + w01 + w02;   // kh == 0
    const float wl0  = w00 + w10 + w20;   // kw == 0
    Bm[(size_t)(c)       * COUT + o] = wsum;
    Bm[(size_t)(64 + c)  * COUT + o] = wk0;
    Bm[(size_t)(128 + c) * COUT + o] = wl0;
    Bm[(size_t)(192 + c) * COUT + o] = w00;
}

// ---------------------------------------------------------------------------
// Kernel 3: 64x64x256 GEMM via V_WMMA_F32_16X16X4_F32 + epilogue.
// 1 block = 512 threads = 16 wave32 waves; each wave owns one 16x16 D tile.
// A layout (16x4, ISA 7.12.2): lane L -> m = L%16; VGPR0/1 = K = 2*(L>=16)+{0,1}
// B layout (4x16): lane L -> n = L%16; VGPR0/1 = K = 2*(L>=16)+{0,1}
// D layout (16x16): VGPR v, lane L -> M = v + 8*(L>=16), N = L%16
// ---------------------------------------------------------------------------
__global__ void __launch_bounds__(512)
wmma_gemm(const float* __restrict__ A, const float* __restrict__ Bm,
          const float* __restrict__ bias, float* __restrict__ out) {
    const int lane = threadIdx.x & 31;
    const int wave = threadIdx.x >> 5;      // 0..15
    const int tm   = (wave >> 2) * 16;      // D tile row base
    const int tn   = (wave & 3) * 16;       // D tile col base
    const int mn   = lane & 15;             // m for A, n for B
    const int half = lane >> 4;             // 0 or 1 (selects K pair)

    v8f acc = {};
#pragma unroll 4
    for (int k0 = 0; k0 < KDIM; k0 += 4) {
        const int ka = k0 + 2 * half;
        v2f a, bv;
        const float* Ap = A + (size_t)(tm + mn) * KDIM + ka;
        a.x = Ap[0];
        a.y = Ap[1];
        bv.x = Bm[(size_t)(ka)     * COUT + tn + mn];
        bv.y = Bm[(size_t)(ka + 1) * COUT + tn + mn];
        // 8 args: (neg_a, A, neg_b, B, c_mod, C, reuse_a, reuse_b)
        acc = __builtin_amdgcn_wmma_f32_16x16x4_f32(
            false, a, false, bv, (short)0, acc, false, false);
    }

    const int n  = tn + mn;
    const float bs = bias[n];
#pragma unroll
    for (int v = 0; v < 8; ++v) {
        const int m = tm + v + 8 * half;
        out[(size_t)m * COUT + n] = (acc[v] * INV_AREA + bs) * MULTV;
    }
}

// ---------------------------------------------------------------------------
extern "C" void kernel_launch(void* const* d_in, const int* in_sizes, int n_in,
                              void* d_out, int out_size, void* d_ws, size_t ws_size,
                              hipStream_t stream) {
    const float* x    = (const float*)d_in[0];   // [64,64,128,128]
    const float* w    = (const float*)d_in[1];   // [64,64,3,3]
    const float* bias = (const float*)d_in[2];   // [64]
    float* out = (float*)d_out;                  // [64,64] (B,Cout,1,1)

    float* A  = (float*)d_ws;                    // 64*256 floats
    float* Bm = A + BATCH * KDIM;                // 256*64 floats

    reduce_planes<<<BATCH * CIN, 256, 0, stream>>>(x, A);
    prep_weights<<<(CIN * COUT + 255) / 256, 256, 0, stream>>>(w, Bm);
    wmma_gemm<<<1, 512, 0, stream>>>(A, Bm, bias, out);
}